// BreakthroughSNNArchitecture_56470230008112
// MI455X (gfx1250) — compile-verified
//
#include <hip/hip_runtime.h>
#include <math.h>

// ---------------- problem constants (from reference) ----------------
#define TSTEPS   128
#define NEURONS  (32 * 8192)          // B*F = 262144
#define NBLOCKS  512
#define NTHREADS 256
#define NPB      512                  // neurons per block
#define NPT      2                    // neurons per thread
#define CHUNK    8                    // timesteps staged per TDM tile
#define NCHUNKS  (TSTEPS / CHUNK)     // 16
#define ROW      9                    // padded LDS row stride in dwords (8 data + 1 pad)

typedef unsigned int u32x4 __attribute__((ext_vector_type(4)));
typedef int          i32x8 __attribute__((ext_vector_type(8)));
typedef int          i32x4 __attribute__((ext_vector_type(4)));

// Low 32 bits of a generic pointer to LDS == byte offset within the LDS aperture.
__device__ __forceinline__ unsigned lds_byte_off(const void* p) {
  return (unsigned)(unsigned long long)p;
}

// Build a 2D-tile TDM descriptor and issue tensor_load_to_lds.
//   tile: tile_cols (dim0, contiguous time) x tile_rows (dim1, neurons)
//   global row stride = stride_elems fp32 elements; LDS rows padded by 1 dword per 8.
__device__ __forceinline__ void tdm_load_tile(const float* gaddr, unsigned lds_off,
                                              unsigned tile_rows, unsigned tile_cols,
                                              unsigned tensor_d0, unsigned tensor_d1,
                                              unsigned stride_elems) {
  unsigned long long ga = (unsigned long long)gaddr;
  u32x4 g0;
  g0.x = 1u;                                          // [1:0] count=1, rest 0
  g0.y = lds_off;                                     // [63:32] lds_addr (bytes)
  g0.z = (unsigned)(ga & 0xFFFFFFFFull);              // global_addr[31:0]
  g0.w = (unsigned)((ga >> 32) & 0x1FFFFFFull)        // global_addr[56:32]
         | (2u << 30);                                // [127:126] type=2 ("image")
  i32x8 g1;
  g1[0] = (int)((2u << 16)                            // data_size: 4 bytes
              | (1u << 20)                            // pad_enable
              | (2u << 22)                            // pad_interval: 8 dwords
              | (0u << 25));                          // pad_amount: 1 dword
  g1[1] = (int)((tensor_d0 & 0xFFFFu) << 16);         // [47:32] bar addr=0 | tensor_dim0 lo16
  g1[2] = (int)((tensor_d0 >> 16) | ((tensor_d1 & 0xFFFFu) << 16));
  g1[3] = (int)((tensor_d1 >> 16) | ((tile_cols & 0xFFFFu) << 16));   // tile_dim0
  g1[4] = (int)(tile_rows & 0xFFFFu);                 // tile_dim1 | tile_dim2=0 (2D)
  g1[5] = (int)stride_elems;                          // tensor_dim0_stride lo32
  g1[6] = 0;                                          // stride hi16 | dim1_stride lo16
  g1[7] = 0;
  i32x4 z4 = {0, 0, 0, 0};
  i32x8 z8 = {0, 0, 0, 0, 0, 0, 0, 0};
  // amdgpu-toolchain (clang-23) 6-arg form: (g0, g1, g2, g3, extra_g, cpol)
  __builtin_amdgcn_tensor_load_to_lds(g0, g1, z4, z4, z8, /*cpol=*/0);
}

__global__ __launch_bounds__(NTHREADS) void snn_lif_persistent(
    const float* __restrict__ in,         // [B,F,T] = [NEURONS, T]
    const float* __restrict__ thr_scale_p,
    const float* __restrict__ a_bias_p,
    float* __restrict__ out,              // spikes | mem_trace | thr_trace | ema | adapt_mean
    float* __restrict__ ws_sums,          // (T+1) x 4 floats, zeroed
    unsigned* __restrict__ ws_cnt)        // (T+1) uints, zeroed
{
  __shared__ float tile[2][NPB * ROW];    // 2 x 18432 B, double-buffered TDM target
  __shared__ float wred[NTHREADS / 32][4];
  __shared__ float bcast;

  const int tid  = threadIdx.x;
  const int wave = tid >> 5;
  const int lane = tid & 31;
  const unsigned blk = blockIdx.x;

  const float thr_scale = thr_scale_p[0];
  const float a_bias    = a_bias_p[0];

  const float alpha = expf(-0.05f);   // dt/tau_mem
  const float beta  = expf(-0.2f);    // dt/tau_syn
  const float gam   = expf(-0.01f);   // dt/tau_adapt
  const float omg   = 1.0f - gam;
  const float invN  = 1.0f / (float)NEURONS;

  float mem[NPT]   = {0.f, 0.f};
  float syn[NPT]   = {0.f, 0.f};
  float adapt[NPT] = {0.f, 0.f};
  float ema = 0.f;

  const float* gbase    = in + (size_t)blk * NPB * TSTEPS;
  float* mem_trace = out + (size_t)NEURONS * TSTEPS;
  float* thr_trace = mem_trace + TSTEPS;

  // ---- prologue: TDM-load chunk 0 (wave 0 only; EXEC ignored by TDM) ----
  if (tid < 32) {
    tdm_load_tile(gbase, lds_byte_off(&tile[0][0]), NPB, CHUNK, TSTEPS, NPB, TSTEPS);
  }

  for (int c = 0; c < NCHUNKS; ++c) {
    const int buf = c & 1;
    if (tid < 32) {
      if (c + 1 < NCHUNKS) {
        tdm_load_tile(gbase + (size_t)(c + 1) * CHUNK,
                      lds_byte_off(&tile[buf ^ 1][0]), NPB, CHUNK, TSTEPS, NPB, TSTEPS);
        __builtin_amdgcn_s_wait_tensorcnt(1);   // chunk c complete (in-order), c+1 in flight
      } else {
        __builtin_amdgcn_s_wait_tensorcnt(0);
      }
    }
    __syncthreads();

    float spkbuf[NPT][CHUNK];

#pragma unroll
    for (int tt = 0; tt < CHUNK; ++tt) {
      const int t = c * CHUNK + tt;
      float s_spk = 0.f, s_mem = 0.f, s_thr = 0.f;
      const float homeo = 0.1f * fmaxf(ema - 0.01f, 0.f);

#pragma unroll
      for (int j = 0; j < NPT; ++j) {
        const int row = tid + j * NTHREADS;
        const float I = tile[buf][row * ROW + tt];
        syn[j]   = beta * syn[j] + I;
        mem[j]   = alpha * mem[j] + syn[j];
        adapt[j] = gam * adapt[j] + omg * mem[j];
        const float thr   = thr_scale + homeo + a_bias + 0.1f * adapt[j];
        const float spike = (mem[j] >= thr) ? 1.f : 0.f;
        mem[j] -= thr * spike;                 // soft reset
        spkbuf[j][tt] = spike;
        s_spk += spike; s_mem += mem[j]; s_thr += thr;
      }

      // wave32 butterfly reduction
#pragma unroll
      for (int off = 16; off > 0; off >>= 1) {
        s_spk += __shfl_xor(s_spk, off, 32);
        s_mem += __shfl_xor(s_mem, off, 32);
        s_thr += __shfl_xor(s_thr, off, 32);
      }
      if (lane == 0) { wred[wave][0] = s_spk; wred[wave][1] = s_mem; wred[wave][2] = s_thr; }
      __syncthreads();

      if (tid == 0) {
        float a = 0.f, b = 0.f, d = 0.f;
#pragma unroll
        for (int w = 0; w < NTHREADS / 32; ++w) { a += wred[w][0]; b += wred[w][1]; d += wred[w][2]; }
        atomicAdd(&ws_sums[t * 4 + 0], a);
        atomicAdd(&ws_sums[t * 4 + 1], b);
        atomicAdd(&ws_sums[t * 4 + 2], d);
        __threadfence();
        atomicAdd(&ws_cnt[t], 1u);
        while (__hip_atomic_load(&ws_cnt[t], __ATOMIC_ACQUIRE, __HIP_MEMORY_SCOPE_AGENT) < NBLOCKS)
          __builtin_amdgcn_s_sleep(1);
        bcast = ws_sums[t * 4 + 0];            // device-wide spike sum for step t
        if (blk == 0) {
          mem_trace[t] = ws_sums[t * 4 + 1] * invN;
          thr_trace[t] = ws_sums[t * 4 + 2] * invN;
        }
      }
      __syncthreads();
      ema = 0.99f * ema + 0.01f * (bcast * invN);   // replicated, deterministic per step
    }

    // write spikes for this chunk: 32 B contiguous per neuron
#pragma unroll
    for (int j = 0; j < NPT; ++j) {
      const size_t ng = (size_t)blk * NPB + tid + j * NTHREADS;
      float4* dst = (float4*)(out + ng * TSTEPS + (size_t)c * CHUNK);
      dst[0] = make_float4(spkbuf[j][0], spkbuf[j][1], spkbuf[j][2], spkbuf[j][3]);
      dst[1] = make_float4(spkbuf[j][4], spkbuf[j][5], spkbuf[j][6], spkbuf[j][7]);
    }
  }

  // ---- final adaptation mean + ema scalars ----
  float s_ad = adapt[0] + adapt[1];
#pragma unroll
  for (int off = 16; off > 0; off >>= 1) s_ad += __shfl_xor(s_ad, off, 32);
  if (lane == 0) wred[wave][0] = s_ad;
  __syncthreads();
  if (tid == 0) {
    float a = 0.f;
#pragma unroll
    for (int w = 0; w < NTHREADS / 32; ++w) a += wred[w][0];
    atomicAdd(&ws_sums[TSTEPS * 4 + 0], a);
    __threadfence();
    atomicAdd(&ws_cnt[TSTEPS], 1u);
    if (blk == 0) {
      while (__hip_atomic_load(&ws_cnt[TSTEPS], __ATOMIC_ACQUIRE, __HIP_MEMORY_SCOPE_AGENT) < NBLOCKS)
        __builtin_amdgcn_s_sleep(1);
      out[(size_t)NEURONS * TSTEPS + 2 * TSTEPS]     = ema;                          // final EMA
      out[(size_t)NEURONS * TSTEPS + 2 * TSTEPS + 1] = ws_sums[TSTEPS * 4] * invN;   // adaptation.mean()
    }
  }
}

extern "C" void kernel_launch(void* const* d_in, const int* in_sizes, int n_in,
                              void* d_out, int out_size, void* d_ws, size_t ws_size,
                              hipStream_t stream) {
  const float* in        = (const float*)d_in[0];
  const float* thr_scale = (const float*)d_in[1];
  const float* a_bias    = (const float*)d_in[2];
  float* out = (float*)d_out;

  float*    ws_sums = (float*)d_ws;
  unsigned* ws_cnt  = (unsigned*)((char*)d_ws + (TSTEPS + 1) * 4 * sizeof(float));
  const size_t zero_bytes = (TSTEPS + 1) * 4 * sizeof(float) + (TSTEPS + 1) * sizeof(unsigned);

  (void)hipMemsetAsync(d_ws, 0, zero_bytes, stream);   // graph-capture legal
  hipLaunchKernelGGL(snn_lif_persistent, dim3(NBLOCKS), dim3(NTHREADS), 0, stream,
                     in, thr_scale, a_bias, out, ws_sums, ws_cnt);
}